// JumpingGCN_19748259627192
// MI455X (gfx1250) — compile-verified
//
#include <hip/hip_runtime.h>
#include <math.h>

typedef __attribute__((ext_vector_type(2))) float v2f;
typedef __attribute__((ext_vector_type(8))) float v8f;

// ---------------------------------------------------------------------------
// Degree / normalization
// ---------------------------------------------------------------------------
__global__ void gcn_deg_init(float* __restrict__ deg, int N) {
    int i = blockIdx.x * blockDim.x + threadIdx.x;
    if (i < N) deg[i] = 1.0f;               // self-loop weight 1
}

__global__ void gcn_deg_edges(float* __restrict__ deg,
                              const int* __restrict__ dst,
                              const float* __restrict__ ew, int E) {
    int e = blockIdx.x * blockDim.x + threadIdx.x;
    if (e < E) unsafeAtomicAdd(&deg[dst[e]], ew[e]);
}

__global__ void gcn_deg_to_dis(float* __restrict__ deg, int N) {
    int i = blockIdx.x * blockDim.x + threadIdx.x;
    if (i < N) deg[i] = rsqrtf(deg[i]);     // deg >= 1 always (self loops)
}

// ---------------------------------------------------------------------------
// FP32 WMMA GEMM:  Out[M,F] = A[M,K] @ W[K,F],  F == NT*16.
// One wave owns one 16-row M-tile and ALL NT column tiles, so the A operand
// (the big streaming input) is read exactly once.  Per K-step of 4:
//   - 1 A-fragment load (2 floats/lane, merges to global_load_b64)
//   - NT B-fragment loads (W is tiny -> L1/L2 resident)
//   - NT independent V_WMMA_F32_16X16X4_F32 (no D->A/B RAW chain between them)
// M % 16 == 0, K % 4 == 0 assumed (50000/512/64/128 all qualify).
// A-frag (16x4, 2 VGPR):  lanes 0-15 -> K base 0, lanes 16-31 -> K base 2.
// B-frag (4x16, 2 VGPR):  VGPR0 rows {0,2}, VGPR1 rows {1,3} across halves.
// D (16x16, 8 VGPR): vgpr v -> row v (lanes 0-15) / row v+8 (lanes 16-31).
// ---------------------------------------------------------------------------
template <int NT>
__global__ void gcn_gemm_wmma_f32(const float* __restrict__ A, int lda,
                                  const float* __restrict__ W,
                                  float* __restrict__ Out,
                                  int K, int mtiles) {
    constexpr int F = NT * 16;
    int wave = blockIdx.x * (blockDim.x >> 5) + (threadIdx.x >> 5);
    if (wave >= mtiles) return;             // whole-wave exit: EXEC stays all-1s
    int lane = threadIdx.x & 31;
    int half = lane >> 4;                   // 0: lanes 0-15, 1: lanes 16-31
    int l    = lane & 15;

    const float* arow  = A + (size_t)(wave * 16 + l) * lda + half * 2;
    const float* wbase = W + (size_t)(half * 2) * F + l;

    v8f acc[NT] = {};
    for (int k = 0; k < K; k += 4) {
        v2f a;
        a.x = arow[k];
        a.y = arow[k + 1];
        const float* wk0 = wbase + (size_t)k * F;
        const float* wk1 = wbase + (size_t)(k + 1) * F;
#pragma unroll
        for (int nt = 0; nt < NT; ++nt) {
            v2f b;
            b.x = wk0[nt * 16];
            b.y = wk1[nt * 16];
            acc[nt] = __builtin_amdgcn_wmma_f32_16x16x4_f32(
                /*neg_a=*/false, a, /*neg_b=*/false, b,
                /*c_mod=*/(short)0, acc[nt], /*reuse_a=*/false, /*reuse_b=*/false);
        }
    }

#pragma unroll
    for (int nt = 0; nt < NT; ++nt) {
        float* obase = Out + (size_t)(wave * 16 + half * 8) * F + nt * 16 + l;
#pragma unroll
        for (int v = 0; v < 8; ++v)
            obase[(size_t)v * F] = acc[nt][v];
    }
}

// ---------------------------------------------------------------------------
// out[row, col0+f] = dis[row]^2 * hpre[row,f] + b[f]   (self-loop msg + bias)
// ---------------------------------------------------------------------------
__global__ void gcn_selfloop_bias(const float* __restrict__ hpre, int F,
                                  const float* __restrict__ dis,
                                  const float* __restrict__ bias,
                                  float* __restrict__ out, int ldo, int col0,
                                  int total) {
    int i = blockIdx.x * blockDim.x + threadIdx.x;
    if (i >= total) return;
    int row = i / F;
    int f   = i - row * F;
    float d = dis[row];
    out[(size_t)row * ldo + col0 + f] = d * d * hpre[(size_t)row * F + f] + bias[f];
}

// ---------------------------------------------------------------------------
// Edge aggregation: one wave per edge (L2-resident scatter, hw f32 atomics).
// out[dst, col0+f] += dis[src]*ew*dis[dst] * hpre[src, f]
// ---------------------------------------------------------------------------
__global__ void gcn_edge_agg(const float* __restrict__ hpre, int F,
                             const int* __restrict__ src,
                             const int* __restrict__ dst,
                             const float* __restrict__ ew,
                             const float* __restrict__ dis,
                             float* __restrict__ out, int ldo, int col0,
                             int E) {
    int e = blockIdx.x * (blockDim.x >> 5) + (threadIdx.x >> 5);
    if (e >= E) return;
    int lane = threadIdx.x & 31;
    int s = src[e];
    int d = dst[e];
    float norm = dis[s] * ew[e] * dis[d];
    const float* hs = hpre + (size_t)s * F;
    float* od = out + (size_t)d * ldo + col0;
    for (int f = lane; f < F; f += 32)
        unsafeAtomicAdd(&od[f], norm * hs[f]);
}

// ---------------------------------------------------------------------------
// In-place softmax over 128 features, one wave per row (wave32: 4/lane).
// ---------------------------------------------------------------------------
__global__ void gcn_softmax128(float* __restrict__ out, int N) {
    int row = blockIdx.x * (blockDim.x >> 5) + (threadIdx.x >> 5);
    if (row >= N) return;
    int lane = threadIdx.x & 31;
    float* p = out + (size_t)row * 128;
    float v0 = p[lane], v1 = p[lane + 32], v2 = p[lane + 64], v3 = p[lane + 96];
    float m = fmaxf(fmaxf(v0, v1), fmaxf(v2, v3));
    for (int off = 16; off; off >>= 1) m = fmaxf(m, __shfl_xor(m, off, 32));
    v0 = __expf(v0 - m); v1 = __expf(v1 - m);
    v2 = __expf(v2 - m); v3 = __expf(v3 - m);
    float s = v0 + v1 + v2 + v3;
    for (int off = 16; off; off >>= 1) s += __shfl_xor(s, off, 32);
    float inv = 1.0f / s;
    p[lane] = v0 * inv; p[lane + 32] = v1 * inv;
    p[lane + 64] = v2 * inv; p[lane + 96] = v3 * inv;
}

// ---------------------------------------------------------------------------
extern "C" void kernel_launch(void* const* d_in, const int* in_sizes, int n_in,
                              void* d_out, int out_size, void* d_ws, size_t ws_size,
                              hipStream_t stream) {
    const float* x   = (const float*)d_in[0];   // [N,512]
    const int*   ei  = (const int*)d_in[1];     // [2,E]
    const float* ea  = (const float*)d_in[2];   // [E]
    const float* W1  = (const float*)d_in[3];   // [512,64]
    const float* b1  = (const float*)d_in[4];
    const float* W2  = (const float*)d_in[5];   // [64,64]
    const float* b2  = (const float*)d_in[6];
    const float* W3  = (const float*)d_in[7];   // [128,128]
    const float* b3  = (const float*)d_in[8];

    const int N = in_sizes[0] / 512;            // 50000 (divisible by 16)
    const int E = in_sizes[2];                  // 800000
    const int* src = ei;
    const int* dst = ei + E;

    // workspace layout
    size_t off = ((size_t)N * sizeof(float) + 255) & ~(size_t)255;
    float* dis  = (float*)d_ws;                                   // [N]
    float* hpre = (float*)((char*)d_ws + off);                    // [N,128] max
    float* h12  = hpre + (size_t)N * 128;                         // [N,128]
    float* outp = (float*)d_out;                                  // [N,128]

    const int TB  = 256;              // 8 waves per block
    const int WPB = TB >> 5;

    // --- normalization: dis = rsqrt(1 + sum_in ew) ---
    gcn_deg_init<<<(N + TB - 1) / TB, TB, 0, stream>>>(dis, N);
    gcn_deg_edges<<<(E + TB - 1) / TB, TB, 0, stream>>>(dis, dst, ea, E);
    gcn_deg_to_dis<<<(N + TB - 1) / TB, TB, 0, stream>>>(dis, N);

    const int mt = N / 16;            // 3125 row tiles
    const int gemmBlocks = (mt + WPB - 1) / WPB;

    // --- layer 1: hpre = x @ W1 [N,64]; h1 -> h12 cols 0..63 ---
    gcn_gemm_wmma_f32<4><<<gemmBlocks, TB, 0, stream>>>(x, 512, W1, hpre, 512, mt);
    gcn_selfloop_bias<<<((N * 64) + TB - 1) / TB, TB, 0, stream>>>(
        hpre, 64, dis, b1, h12, 128, 0, N * 64);
    gcn_edge_agg<<<(E + WPB - 1) / WPB, TB, 0, stream>>>(
        hpre, 64, src, dst, ea, dis, h12, 128, 0, E);

    // --- layer 2: hpre = h1 @ W2 [N,64]; h2 -> h12 cols 64..127 ---
    gcn_gemm_wmma_f32<4><<<gemmBlocks, TB, 0, stream>>>(
        h12 /*cols 0..63*/, 128, W2, hpre, 64, mt);
    gcn_selfloop_bias<<<((N * 64) + TB - 1) / TB, TB, 0, stream>>>(
        hpre, 64, dis, b2, h12, 128, 64, N * 64);
    gcn_edge_agg<<<(E + WPB - 1) / WPB, TB, 0, stream>>>(
        hpre, 64, src, dst, ea, dis, h12, 128, 64, E);

    // --- layer 3: hpre = [h1|h2] @ W3 [N,128]; aggregate into d_out ---
    gcn_gemm_wmma_f32<8><<<gemmBlocks, TB, 0, stream>>>(
        h12, 128, W3, hpre, 128, mt);
    gcn_selfloop_bias<<<((N * 128) + TB - 1) / TB, TB, 0, stream>>>(
        hpre, 128, dis, b3, outp, 128, 0, N * 128);
    gcn_edge_agg<<<(E + WPB - 1) / WPB, TB, 0, stream>>>(
        hpre, 128, src, dst, ea, dis, outp, 128, 0, E);

    // --- softmax over 128 features ---
    gcn_softmax128<<<(N + WPB - 1) / WPB, TB, 0, stream>>>(outp, N);
}